// TexturePointNet_17085379904243
// MI455X (gfx1250) — compile-verified
//
#include <hip/hip_runtime.h>
#include <hip/hip_bf16.h>

typedef float v2f __attribute__((ext_vector_type(2)));
typedef float v8f __attribute__((ext_vector_type(8)));

#define BN_INV 0.9999950000374997f   // 1/sqrt(1+1e-5)

// ---------------------------------------------------------------------------
// Farthest point sampling: one workgroup per batch. dist[] lives in global
// scratch (B*N floats). Emits sampled centers directly (gather fused).
// ---------------------------------------------------------------------------
__global__ void fps_kernel(const float* __restrict__ xyz, float* __restrict__ dist,
                           int N, int npoint, float* __restrict__ centers)
{
    const int b = blockIdx.x;
    const float* p = xyz + (size_t)b * N * 2;
    float* d = dist + (size_t)b * N;
    __shared__ float rv[512];
    __shared__ int   ri[512];
    __shared__ int   s_far;
    const int tid = threadIdx.x;

    for (int i = tid; i < N; i += blockDim.x) d[i] = 1e10f;
    if (tid == 0) s_far = 0;
    __syncthreads();

    for (int m = 0; m < npoint; ++m) {
        const int far = s_far;
        const float cx = p[far * 2 + 0];
        const float cy = p[far * 2 + 1];
        if (tid == 0) {
            centers[((size_t)b * npoint + m) * 2 + 0] = cx;
            centers[((size_t)b * npoint + m) * 2 + 1] = cy;
        }
        float bestv = -1.0f; int besti = 0;
        for (int i = tid; i < N; i += blockDim.x) {
            const float dx = p[i * 2 + 0] - cx;
            const float dy = p[i * 2 + 1] - cy;
            const float dd = dx * dx + dy * dy;
            const float nd = fminf(d[i], dd);
            d[i] = nd;
            if (nd > bestv) { bestv = nd; besti = i; }
        }
        rv[tid] = bestv; ri[tid] = besti;
        __syncthreads();
        for (int s = blockDim.x >> 1; s > 0; s >>= 1) {
            if (tid < s) {
                const float ov = rv[tid + s]; const int oi = ri[tid + s];
                if (ov > rv[tid] || (ov == rv[tid] && oi < ri[tid])) { rv[tid] = ov; ri[tid] = oi; }
            }
            __syncthreads();
        }
        if (tid == 0) s_far = ri[0];
        __syncthreads();
    }
}

// ---------------------------------------------------------------------------
// Ball query: one thread per centroid; ordered scan with early exit == the
// reference's "mask, sort, take first nsample, backfill with first".
// ---------------------------------------------------------------------------
__global__ void ball_query_kernel(const float* __restrict__ xyz,
                                  const float* __restrict__ centers,
                                  int* __restrict__ out_idx,
                                  int N, int S, int nsample, float r2, int B)
{
    const int t = blockIdx.x * blockDim.x + threadIdx.x;
    if (t >= B * S) return;
    const int b = t / S;
    const float* p = xyz + (size_t)b * N * 2;
    const float cx = centers[(size_t)t * 2 + 0];
    const float cy = centers[(size_t)t * 2 + 1];
    int* out = out_idx + (size_t)t * nsample;
    int cnt = 0, first = 0;
    for (int i = 0; i < N; ++i) {
        const float dx = p[i * 2 + 0] - cx;
        const float dy = p[i * 2 + 1] - cy;
        if (dx * dx + dy * dy <= r2) {
            if (cnt == 0) first = i;
            out[cnt++] = i;
            if (cnt == nsample) break;
        }
    }
    for (; cnt < nsample; ++cnt) out[cnt] = first;
}

// ---------------------------------------------------------------------------
// Per-layer preprocessing: transpose + zero-pad weights to Wt[coutPad][kpad]
// (kpad = ceil4(cin)), and fold BN into per-channel alpha/beta (padded zeros).
// Makes the WMMA inner loop completely branchless.
// ---------------------------------------------------------------------------
__global__ void prep_layer_kernel(const float* __restrict__ W, const float* __restrict__ b,
                                  const float* __restrict__ g, const float* __restrict__ e,
                                  int cin, int cout, int kpad, int coutPad,
                                  float* __restrict__ Wt, float* __restrict__ al,
                                  float* __restrict__ bt)
{
    const int i = blockIdx.x * blockDim.x + threadIdx.x;
    const int total = coutPad * kpad;
    if (i < total) {
        const int c = i / kpad, k = i - c * kpad;
        Wt[i] = (c < cout && k < cin) ? W[(size_t)k * cout + c] : 0.0f;
    }
    if (i < coutPad) {
        float a = 0.0f, bb = 0.0f;
        if (i < cout) { a = g[i] * BN_INV; bb = b[i] * a + e[i]; }
        al[i] = a; bt[i] = bb;
    }
}

// ---------------------------------------------------------------------------
// Fused 4-layer pointwise MLP + max-pool, one workgroup per group.
// All GEMMs on V_WMMA_F32_16X16X4_F32 (exact fp32). Activations ping-pong in
// LDS (channel-padded to 16 => zero K-edges). Branchless K loop:
//   A frag: ds_load_b64 from LDS;  B frag: global_load_b64 from Wt (transposed).
// Last layer max-pools from accumulators via LDS bit-atomicMax (ReLU => >=0).
// ---------------------------------------------------------------------------
struct MlpP {
    const float* Wt[4];   // [coutPad][kpad] transposed, zero-padded
    const float* al[4];   // alpha per padded out-channel
    const float* bt[4];   // beta  per padded out-channel
    int kpad[4];          // ceil4(cin)
    int coutPad[4];       // ceil16(cout)
};

__global__ void mlp_max_kernel(MlpP P, int mode,
                               const float* __restrict__ src,
                               const float* __restrict__ centers,
                               const int* __restrict__ idx,
                               int Nsrc, int M, int S, int apad, int cin0,
                               float* __restrict__ out)
{
    extern __shared__ float smem[];
    float* buf0 = smem;
    float* buf1 = smem + (size_t)M * apad;
    float* pool = buf1 + (size_t)M * apad;

    const int g = blockIdx.x;
    const int b = g / S;
    const int tid  = threadIdx.x;
    const int lane = tid & 31;
    const int hh   = lane >> 4;      // lane half (K-pair select, ISA 7.12.2)
    const int lr   = lane & 15;
    const int wave = tid >> 5;
    const int nwaves = blockDim.x >> 5;

    // ---- Stage activations (gather variants) into buf0, zero-padded ----
    const int cpad0 = (cin0 + 15) & ~15;
    for (int e = tid; e < M * cpad0; e += blockDim.x) {
        const int j = e / cpad0, c = e - j * cpad0;
        float v = 0.0f;
        if (c < cin0) {
            if (mode == 0) {            // SA1: xyz gather minus center
                const int pi = idx[(size_t)g * M + j];
                v = src[((size_t)b * Nsrc + pi) * 2 + c] - centers[(size_t)g * 2 + c];
            } else if (mode == 1) {     // SA2: feature gather
                const int pi = idx[(size_t)g * M + j];
                v = src[((size_t)b * Nsrc + pi) * cin0 + c];
            } else {                    // SA3: direct rows
                v = src[((size_t)g * M + j) * cin0 + c];
            }
        }
        buf0[j * cpad0 + c] = v;
    }
    const int coutL = P.coutPad[3];            // == actual cout for all 3 MLPs
    for (int c = tid; c < coutL; c += blockDim.x) pool[c] = 0.0f;
    __syncthreads();

    float* cur = buf0; float* nxt = buf1;
    int curStride = cpad0;

    for (int l = 0; l < 4; ++l) {
        const int kpad = P.kpad[l];
        const int coutPad = P.coutPad[l];
        const int ntn = coutPad >> 4;
        const int ntm = M >> 4;
        const int tiles = ntm * ntn;
        const float* __restrict__ Wt = P.Wt[l];
        const float* __restrict__ al = P.al[l];
        const float* __restrict__ bt = P.bt[l];
        const bool last = (l == 3);

        for (int t = wave; t < tiles; t += nwaves) {
            const int mt = t / ntn, nt = t - mt * ntn;
            const int m0 = mt << 4, n0 = nt << 4;
            const int row = m0 + lr;
            const int col = n0 + lr;
            const float* __restrict__ arow = cur + row * curStride;
            const float* __restrict__ wrow = Wt + (size_t)col * kpad;
            v8f acc = {0.f, 0.f, 0.f, 0.f, 0.f, 0.f, 0.f, 0.f};

            for (int kk = 0; kk < kpad; kk += 4) {
                const int k0 = kk + 2 * hh;
                const v2f a  = *(const v2f*)(arow + k0);   // LDS b64, 8B aligned
                const v2f bb = *(const v2f*)(wrow + k0);   // global b64, 8B aligned
                acc = __builtin_amdgcn_wmma_f32_16x16x4_f32(
                    /*neg_a=*/false, a, /*neg_b=*/false, bb,
                    /*c_mod=*/(short)0, acc, /*reuse_a=*/false, /*reuse_b=*/false);
            }

            const float alpha = al[col];
            const float beta  = bt[col];
            if (!last) {
#pragma unroll
                for (int r = 0; r < 8; ++r) {
                    const float y = fmaxf(acc[r] * alpha + beta, 0.0f);
                    const int rrow = m0 + r + (hh << 3);   // D rows per ISA layout
                    nxt[rrow * coutPad + col] = y;
                }
            } else {
                float mx = 0.0f;
#pragma unroll
                for (int r = 0; r < 8; ++r)
                    mx = fmaxf(mx, fmaxf(acc[r] * alpha + beta, 0.0f));
                atomicMax((int*)&pool[col], __float_as_int(mx)); // mx >= 0
            }
        }
        __syncthreads();
        float* tmp = cur; cur = nxt; nxt = tmp;
        curStride = coutPad;
    }

    for (int c = tid; c < coutL; c += blockDim.x)
        out[(size_t)g * coutL + c] = pool[c];
}

// ---------------------------------------------------------------------------
// FC head: 32x512 -> 256 -> 128 -> 3 (tiny; single workgroup, plain VALU).
// ---------------------------------------------------------------------------
__global__ void fc_kernel(const float* __restrict__ l3,
                          const float* W1, const float* b1, const float* g1, const float* e1,
                          const float* W2, const float* b2, const float* g2, const float* e2,
                          const float* W3, const float* b3,
                          float* __restrict__ out, int B)
{
    __shared__ float h1[32 * 256];
    __shared__ float h2[32 * 128];
    const int tid = threadIdx.x;

    for (int e = tid; e < B * 256; e += blockDim.x) {
        const int r = e >> 8, c = e & 255;
        const float* x = l3 + (size_t)r * 512;
        float s = 0.0f;
        for (int k = 0; k < 512; ++k) s += x[k] * W1[(size_t)k * 256 + c];
        s = (s + b1[c]) * (g1[c] * BN_INV) + e1[c];
        h1[e] = fmaxf(s, 0.0f);
    }
    __syncthreads();
    for (int e = tid; e < B * 128; e += blockDim.x) {
        const int r = e >> 7, c = e & 127;
        const float* x = h1 + r * 256;
        float s = 0.0f;
        for (int k = 0; k < 256; ++k) s += x[k] * W2[(size_t)k * 128 + c];
        s = (s + b2[c]) * (g2[c] * BN_INV) + e2[c];
        h2[e] = fmaxf(s, 0.0f);
    }
    __syncthreads();
    for (int e = tid; e < B * 3; e += blockDim.x) {
        const int r = e / 3, c = e - r * 3;
        const float* x = h2 + r * 128;
        float s = 0.0f;
        for (int k = 0; k < 128; ++k) s += x[k] * W3[(size_t)k * 3 + c];
        out[e] = s + b3[c];
    }
}

// ---------------------------------------------------------------------------
extern "C" void kernel_launch(void* const* d_in, const int* in_sizes, int n_in,
                              void* d_out, int out_size, void* d_ws, size_t ws_size,
                              hipStream_t stream)
{
    (void)in_sizes; (void)n_in; (void)out_size; (void)ws_size;
    const int B = 32, N = 16384;
    auto F = [&](int i) { return (const float*)d_in[i]; };

    // Workspace carve-out (256B aligned).
    char* wsb = (char*)d_ws;
    size_t off = 0;
    auto alloc = [&](size_t bytes) {
        void* p = wsb + off;
        off = (off + bytes + 255) & ~(size_t)255;
        return p;
    };
    float* dist   = (float*)alloc((size_t)B * N * 4);        // FPS scratch
    float* l1_xyz = (float*)alloc((size_t)B * 128 * 2 * 4);
    int*   idx1   = (int*)  alloc((size_t)B * 128 * 32 * 4);
    float* l1     = (float*)alloc((size_t)B * 128 * 64 * 4);
    float* l2_xyz = (float*)alloc((size_t)B * 64 * 2 * 4);
    int*   idx2   = (int*)  alloc((size_t)B * 64 * 32 * 4);
    float* l2     = (float*)alloc((size_t)B * 64 * 128 * 4);
    float* l3     = (float*)alloc((size_t)B * 512 * 4);

    // Parameter packs: d_in = [x, sa1(16), sa2(16), sa3(16), fc(10)].
    const int ci[3][4] = {{2, 2, 32, 32}, {64, 64, 64, 64}, {128, 128, 128, 256}};
    const int co[3][4] = {{2, 32, 32, 64}, {64, 64, 64, 128}, {128, 128, 256, 512}};
    const int base[3]  = {1, 17, 33};

    MlpP P[3];
    for (int m = 0; m < 3; ++m) {
        for (int l = 0; l < 4; ++l) {
            const int cin = ci[m][l], cout = co[m][l];
            const int kpad = (cin + 3) & ~3;
            const int coutPad = (cout + 15) & ~15;
            float* Wt = (float*)alloc((size_t)coutPad * kpad * 4);
            float* al = (float*)alloc((size_t)coutPad * 4);
            float* bt = (float*)alloc((size_t)coutPad * 4);
            P[m].Wt[l] = Wt; P[m].al[l] = al; P[m].bt[l] = bt;
            P[m].kpad[l] = kpad; P[m].coutPad[l] = coutPad;
            const int idx0 = base[m] + 4 * l;
            const int total = coutPad * kpad;
            prep_layer_kernel<<<(total + 255) / 256, 256, 0, stream>>>(
                F(idx0), F(idx0 + 1), F(idx0 + 2), F(idx0 + 3),
                cin, cout, kpad, coutPad, Wt, al, bt);
        }
    }
    const float* x = F(0);

    // SA3 needs 133120 B of dynamic LDS (320 KB/WGP on CDNA5 permits it).
    hipFuncSetAttribute((const void*)mlp_max_kernel,
                        hipFuncAttributeMaxDynamicSharedMemorySize, 133120);

    // ---- SA1 ----
    fps_kernel<<<B, 512, 0, stream>>>(x, dist, N, 128, l1_xyz);
    ball_query_kernel<<<(B * 128 + 255) / 256, 256, 0, stream>>>(
        x, l1_xyz, idx1, N, 128, 32, 0.2f * 0.2f, B);
    mlp_max_kernel<<<B * 128, 256, (2 * 32 * 32 + 64) * 4, stream>>>(
        P[0], 0, x, l1_xyz, idx1, N, 32, 128, 32, 2, l1);

    // ---- SA2 ----
    fps_kernel<<<B, 512, 0, stream>>>(l1_xyz, dist, 128, 64, l2_xyz);
    ball_query_kernel<<<(B * 64 + 255) / 256, 256, 0, stream>>>(
        l1_xyz, l2_xyz, idx2, 128, 64, 32, 0.4f * 0.4f, B);
    mlp_max_kernel<<<B * 64, 256, (2 * 32 * 64 + 128) * 4, stream>>>(
        P[1], 1, l1, nullptr, idx2, 128, 32, 64, 64, 64, l2);

    // ---- SA3 (whole 64-point set as one group per batch) ----
    mlp_max_kernel<<<B, 256, (2 * 64 * 256 + 512) * 4, stream>>>(
        P[2], 2, l2, nullptr, nullptr, 64, 64, 1, 256, 128, l3);

    // ---- FC head ----
    fc_kernel<<<1, 256, 0, stream>>>(l3,
        F(49), F(50), F(51), F(52),
        F(53), F(54), F(55), F(56),
        F(57), F(58), (float*)d_out, B);
}